// EdgeWeightedSumAndMax_6373731467766
// MI455X (gfx1250) — compile-verified
//
#include <hip/hip_runtime.h>
#include <math.h>

typedef __attribute__((ext_vector_type(2))) float v2f;
typedef __attribute__((ext_vector_type(8))) float v8f;

#define IN_F 128
#define ROW_STRIDE 132   // 128 + 4 floats pad -> conflict-free ds_load_b64 of A fragments
#define WPB 4            // waves per block
#define GROUP 16         // edges per wave iteration

__device__ __forceinline__ unsigned enc_f32(float f) {
  unsigned b = __float_as_uint(f);
  return (b & 0x80000000u) ? ~b : (b | 0x80000000u);
}

// Initialize output: sum slots (cols 0..127) = 0.0f (bits 0), max slots (cols 128..255)
// = encoded(-inf) = 0x007FFFFF for the integer atomicMax staging.
__global__ void eswm_init_kernel(unsigned* __restrict__ out_u, int total) {
  int i = blockIdx.x * blockDim.x + threadIdx.x;
  if (i < total) out_u[i] = ((i & 255) < 128) ? 0u : 0x007FFFFFu;
}

// Decode max slots in place: monotone uint -> float bits.
__global__ void eswm_decode_kernel(unsigned* __restrict__ out_u, int gmax) {
  int i = blockIdx.x * blockDim.x + threadIdx.x;
  if (i < gmax) {
    int pos = (i >> 7) * 256 + 128 + (i & 127);
    unsigned u = out_u[pos];
    out_u[pos] = (u & 0x80000000u) ? (u & 0x7FFFFFFFu) : ~u;
  }
}

__global__ __launch_bounds__(WPB * 32) void eswm_main_kernel(
    const float* __restrict__ x, const int* __restrict__ seg,
    const float* __restrict__ Wg, const float* __restrict__ bg,
    float* __restrict__ out_f, unsigned* __restrict__ out_u,
    int E, int nwaves) {
  __shared__ __attribute__((aligned(16))) float lds[WPB * GROUP * ROW_STRIDE + IN_F];
  const int tid = threadIdx.x;
  const int wbase = WPB * GROUP * ROW_STRIDE;  // W copy lives after the tiles
  lds[wbase + tid] = Wg[tid];                  // blockDim == 128 == IN_F
  __syncthreads();

  const int lane = tid & 31;
  const int wave = tid >> 5;
  const int tbase = wave * GROUP * ROW_STRIDE;
  const int gw = blockIdx.x * WPB + wave;
  const int per = (E + nwaves - 1) / nwaves;
  int e0 = gw * per;
  int eend = e0 + per;
  if (eend > E) eend = E;
  if (e0 >= eend) return;

  const float bias = bg[0];
  const int half = lane >> 4;   // which half-wave (K 0,1 vs K 2,3 of A; K0/K2 vs K1/K3 rows of B)
  const int m = lane & 15;      // edge row within group for A fragments
  const float4* __restrict__ x4 = (const float4*)x;

  float s0 = 0.f, s1 = 0.f, s2 = 0.f, s3 = 0.f;
  float m0 = -INFINITY, m1 = -INFINITY, m2 = -INFINITY, m3 = -INFINITY;
  int cur = seg[e0];

  auto flush = [&](int g) {
    float* po = out_f + (size_t)g * 256 + lane * 4;
    atomicAdd(po + 0, s0); atomicAdd(po + 1, s1);
    atomicAdd(po + 2, s2); atomicAdd(po + 3, s3);
    unsigned* pu = out_u + (size_t)g * 256 + 128 + lane * 4;
    atomicMax(pu + 0, enc_f32(m0)); atomicMax(pu + 1, enc_f32(m1));
    atomicMax(pu + 2, enc_f32(m2)); atomicMax(pu + 3, enc_f32(m3));
    s0 = s1 = s2 = s3 = 0.f;
    m0 = m1 = m2 = m3 = -INFINITY;
  };

  while (e0 < eend) {
    int n = eend - e0;
    if (n > GROUP) n = GROUP;
    // lane i (i<16) holds segment id of edge e0+i (clamped for ragged tail)
    int sid = seg[e0 + (m < n ? m : n - 1)];

    // Stage 16x128 tile: coalesced b128 loads, conflict-free b128 LDS stores.
    for (int i = 0; i < n; ++i) {
      float4 v = x4[(e0 + i) * (IN_F / 4) + lane];
      *(float4*)&lds[tbase + i * ROW_STRIDE + lane * 4] = v;
    }

    // Gate GEMV via WMMA: D[m][n] = dot(edge_m, W) for every n (B replicated along N).
    // A (16x4 f32): lanes 0-15 VGPR{0,1} = K{0,1}; lanes 16-31 = K{2,3}.
    // B (4x16 f32): VGPR0 = rows K0 (lanes 0-15) / K2 (lanes 16-31); VGPR1 = K1 / K3.
    v8f acc = {};
#pragma unroll
    for (int c = 0; c < 32; ++c) {
      v2f a = *(const v2f*)&lds[tbase + m * ROW_STRIDE + 4 * c + 2 * half];
      v2f bv = *(const v2f*)&lds[wbase + 4 * c + 2 * half];
      acc = __builtin_amdgcn_wmma_f32_16x16x4_f32(false, a, false, bv,
                                                  (short)0, acc, false, false);
    }
    // D layout: VGPR j -> edge j (lanes 0-15) / edge j+8 (lanes 16-31), replicated over N.
    float t[8];
#pragma unroll
    for (int j = 0; j < 8; ++j) t[j] = tanhf(acc[j] + bias);

#pragma unroll
    for (int i = 0; i < GROUP; ++i) {
      if (i >= n) break;                 // n is wave-uniform
      int sid_i = __shfl(sid, i);
      if (sid_i != cur) { flush(cur); cur = sid_i; }
      float g = __shfl(t[i & 7], (i < 8) ? 0 : 16);
      int o = tbase + i * ROW_STRIDE + lane * 4;
      float vx = lds[o + 0], vy = lds[o + 1], vz = lds[o + 2], vw = lds[o + 3];
      s0 = fmaf(g, vx, s0); s1 = fmaf(g, vy, s1);
      s2 = fmaf(g, vz, s2); s3 = fmaf(g, vw, s3);
      m0 = fmaxf(m0, vx); m1 = fmaxf(m1, vy);
      m2 = fmaxf(m2, vz); m3 = fmaxf(m3, vw);
    }
    e0 += n;
  }
  flush(cur);
}

extern "C" void kernel_launch(void* const* d_in, const int* in_sizes, int n_in,
                              void* d_out, int out_size, void* d_ws, size_t ws_size,
                              hipStream_t stream) {
  (void)n_in; (void)d_ws; (void)ws_size;
  const float* x = (const float*)d_in[0];
  const int* seg = (const int*)d_in[1];
  // d_in[2] = num_graphs (scalar) — G recovered from out_size instead
  const float* W = (const float*)d_in[3];
  const float* b = (const float*)d_in[4];
  const int E = in_sizes[1];
  const int G = out_size / 256;

  eswm_init_kernel<<<(out_size + 255) / 256, 256, 0, stream>>>(
      (unsigned*)d_out, out_size);

  const int blocks = 2048;
  eswm_main_kernel<<<blocks, WPB * 32, 0, stream>>>(
      x, seg, W, b, (float*)d_out, (unsigned*)d_out, E, blocks * WPB);

  const int gmax = G * 128;
  eswm_decode_kernel<<<(gmax + 255) / 256, 256, 0, stream>>>(
      (unsigned*)d_out, gmax);
}